// NeuralMap_27238682591928
// MI455X (gfx1250) — compile-verified
//
#include <hip/hip_runtime.h>
#include <math.h>

// ---------------------------------------------------------------------------
// Problem constants (from reference): Z [512,256], SOM [32,32,256] -> L=1024
// Outputs: kl[512] then delta[512*1024*256] concatenated in d_out.
// ---------------------------------------------------------------------------
#define BATCH 512
#define LAT   1024      // 32*32 lattice nodes
#define EMB   256

typedef __attribute__((ext_vector_type(2))) float v2f;
typedef __attribute__((ext_vector_type(4))) float v4f;
typedef __attribute__((ext_vector_type(8))) float v8f;

__device__ __forceinline__ v8f wmma4(v2f a, v2f b, v8f c) {
  // V_WMMA_F32_16X16X4_F32 : D = A(16x4) * B(4x16) + C, full f32
  return __builtin_amdgcn_wmma_f32_16x16x4_f32(false, a, false, b, (short)0, c,
                                               false, false);
}

__device__ __forceinline__ float wred_add(float v) {
#pragma unroll
  for (int o = 16; o > 0; o >>= 1) v += __shfl_xor(v, o, 32);
  return v;  // xor butterfly: every lane holds the total
}

// ---------------------------------------------------------------------------
// K0: scalar params from step:  eta, sigma, eta/B
// ---------------------------------------------------------------------------
__global__ void params_kernel(const int* __restrict__ step, float* __restrict__ p) {
  float n = (float)step[0];
  float decay = expf(-n * (1.0f / 100.0f));  // ALPHA = 100
  p[0] = 0.3f * decay;                        // eta
  p[1] = 16.0f * decay;                       // sigma
  p[2] = p[0] * (1.0f / (float)BATCH);        // eta / B
}

// ---------------------------------------------------------------------------
// Row squared-norms: rows x 256, one wave per row
// ---------------------------------------------------------------------------
__global__ void rownorm_kernel(const float* __restrict__ in, float* __restrict__ out,
                               int rows) {
  int wave = (int)(blockIdx.x * blockDim.x + threadIdx.x) >> 5;
  int lane = threadIdx.x & 31;
  if (wave >= rows) return;
  const float* p = in + (size_t)wave * EMB;
  float s = 0.0f;
#pragma unroll
  for (int c = lane; c < EMB; c += 32) { float v = p[c]; s += v * v; }
  s = wred_add(s);
  if (lane == 0) out[wave] = s;
}

// ---------------------------------------------------------------------------
// WMMA accumulate: C += A(16 rows of Arow, K) * Brow^T tile, both row-major
// A row m: Arow + m*K (contiguous K).  B[k,n] = Brow[n*K + k].
// ---------------------------------------------------------------------------
__device__ __forceinline__ v8f gemm_tile_rowmajor(const float* __restrict__ A,
                                                  const float* __restrict__ Bt,
                                                  int m0, int n0, int lane, int K) {
  int r  = lane & 15;
  int kb = (lane >> 4) << 1;  // lanes 0-15 -> K {0,1}; lanes 16-31 -> K {2,3}
  const float* ap = A  + (size_t)(m0 + r) * K + kb;
  const float* bp = Bt + (size_t)(n0 + r) * K + kb;
  v8f acc = {0.f, 0.f, 0.f, 0.f, 0.f, 0.f, 0.f, 0.f};
#pragma unroll 8
  for (int k0 = 0; k0 < K; k0 += 4) {
    v2f a = *(const v2f*)(ap + k0);
    v2f b = *(const v2f*)(bp + k0);
    acc = wmma4(a, b, acc);
  }
  return acc;
}

// ---------------------------------------------------------------------------
// K1: score[b,j] = Snorm[j] - 2 * (Z . SOM^T)[b,j]   (argmin-equivalent dist)
// One wave per 16x16 tile; grid covers exactly (512/16)*(1024/16) tiles.
// ---------------------------------------------------------------------------
__global__ void gemm_score_kernel(const float* __restrict__ Z,
                                  const float* __restrict__ SOM,
                                  const float* __restrict__ snorm,
                                  float* __restrict__ Dout) {
  int wave = (int)(blockIdx.x * blockDim.x + threadIdx.x) >> 5;
  int lane = threadIdx.x & 31;
  int tn = wave & ((LAT / 16) - 1);
  int tm = wave >> 6;                 // LAT/16 = 64
  int m0 = tm * 16, n0 = tn * 16;
  v8f acc = gemm_tile_rowmajor(Z, SOM, m0, n0, lane, EMB);
  int n  = n0 + (lane & 15);
  int mb = m0 + ((lane >> 4) << 3);
  float sn = snorm[n];
#pragma unroll
  for (int i = 0; i < 8; ++i)
    Dout[(size_t)(mb + i) * LAT + n] = sn - 2.0f * acc[i];
}

// ---------------------------------------------------------------------------
// K2: argmin over each row of score -> BMU flat index (first-min tie-break)
// ---------------------------------------------------------------------------
__global__ void argmin_kernel(const float* __restrict__ D, int* __restrict__ U) {
  int b = (int)(blockIdx.x * blockDim.x + threadIdx.x) >> 5;
  int lane = threadIdx.x & 31;
  if (b >= BATCH) return;
  float mv = 3.4e38f;
  int   mi = 0;
  const float* row = D + (size_t)b * LAT;
#pragma unroll
  for (int j = lane; j < LAT; j += 32) {
    float v = row[j];
    if (v < mv) { mv = v; mi = j; }   // increasing j => keeps first min per lane
  }
#pragma unroll
  for (int o = 16; o > 0; o >>= 1) {
    float v2 = __shfl_xor(mv, o, 32);
    int   i2 = __shfl_xor(mi, o, 32);
    if (v2 < mv || (v2 == mv && i2 < mi)) { mv = v2; mi = i2; }
  }
  if (lane == 0) U[b] = mi;
}

// ---------------------------------------------------------------------------
// K3: neighborhood table h[b,j] = exp(-((ui-i)^2+(uj-j)^2)/(2 sigma^2))
// ---------------------------------------------------------------------------
__global__ void hbuf_kernel(const int* __restrict__ U, const float* __restrict__ prm,
                            float* __restrict__ Hb) {
  int idx = blockIdx.x * blockDim.x + threadIdx.x;  // BATCH*LAT
  int b = idx >> 10, j = idx & (LAT - 1);
  int u = U[b];
  float di = (float)((u >> 5) - (j >> 5));
  float dj = (float)((u & 31) - (j & 31));
  float sigma = prm[1];
  Hb[idx] = expf(-(di * di + dj * dj) / (2.0f * sigma * sigma));
}

// ---------------------------------------------------------------------------
// K4: the bandwidth kernel. delta[b,j,e] = eta*h[b,j]*(Z[b,e]-SOM[j,e])
// 128-bit lanes; 536 MB of streaming stores -> HBM-bound (~23 us at 23.3 TB/s).
// delta is never read back (mean recomputed via H^T Z GEMM), so stores use
// the CDNA5 non-temporal store policy (TH=NT) to avoid sweeping the 192 MB L2
// and evicting the hot Z/SOM/Hb working set. Loads stay temporal (reused).
// ---------------------------------------------------------------------------
__global__ void delta_kernel(const float* __restrict__ Z, const float* __restrict__ SOM,
                             const float* __restrict__ Hb, const float* __restrict__ prm,
                             float* __restrict__ delta) {
  int idx = blockIdx.x * blockDim.x + threadIdx.x;   // B*L*E/4 = 33,554,432
  int e4 = idx & 63;
  int j  = (idx >> 6) & (LAT - 1);
  int b  = idx >> 16;
  float scale = prm[0] * Hb[b * LAT + j];
  v4f z = ((const v4f*)Z)[b * 64 + e4];
  v4f s = ((const v4f*)SOM)[j * 64 + e4];
  v4f o = scale * (z - s);
  __builtin_nontemporal_store(o, (v4f*)delta + idx);
}

// ---------------------------------------------------------------------------
// K5: Hsum[j] = sum_b h[b,j]
// ---------------------------------------------------------------------------
__global__ void hsum_kernel(const float* __restrict__ Hb, float* __restrict__ out) {
  int j = (int)(blockIdx.x * blockDim.x + threadIdx.x) >> 5;
  int lane = threadIdx.x & 31;
  if (j >= LAT) return;
  float s = 0.0f;
#pragma unroll
  for (int b = lane; b < BATCH; b += 32) s += Hb[(size_t)b * LAT + j];
  s = wred_add(s);
  if (lane == 0) out[j] = s;
}

// ---------------------------------------------------------------------------
// K6: SOM_new[j,e] = SOM[j,e] + (eta/B)*( (H^T Z)[j,e] - SOM[j,e]*Hsum[j] )
// WMMA GEMM: M=LAT (j), N=EMB (e), K=BATCH (b); A[m,k]=Hb[k*LAT+m]
// ---------------------------------------------------------------------------
__global__ void gemm_hz_kernel(const float* __restrict__ Hb, const float* __restrict__ Z,
                               const float* __restrict__ SOM, const float* __restrict__ hsum,
                               const float* __restrict__ prm, float* __restrict__ SOMnew) {
  int wave = (int)(blockIdx.x * blockDim.x + threadIdx.x) >> 5;
  int lane = threadIdx.x & 31;
  int tn = wave & 15;                 // EMB/16 = 16
  int tm = wave >> 4;
  int m0 = tm * 16, n0 = tn * 16;
  int r  = lane & 15;
  int kb = (lane >> 4) << 1;
  v8f acc = {0.f, 0.f, 0.f, 0.f, 0.f, 0.f, 0.f, 0.f};
#pragma unroll 4
  for (int k0 = 0; k0 < BATCH; k0 += 4) {
    v2f a, b;
    a.x = Hb[(size_t)(k0 + kb) * LAT + (m0 + r)];
    a.y = Hb[(size_t)(k0 + kb + 1) * LAT + (m0 + r)];
    b.x = Z[(size_t)(k0 + kb) * EMB + (n0 + r)];
    b.y = Z[(size_t)(k0 + kb + 1) * EMB + (n0 + r)];
    acc = wmma4(a, b, acc);
  }
  float c1 = prm[2];                  // eta / B
  int e  = n0 + (lane & 15);
  int jb = m0 + ((lane >> 4) << 3);
#pragma unroll
  for (int i = 0; i < 8; ++i) {
    int j = jb + i;
    float s = SOM[(size_t)j * EMB + e];
    SOMnew[(size_t)j * EMB + e] = s + c1 * (acc[i] - s * hsum[j]);
  }
}

// ---------------------------------------------------------------------------
// K7: Qraw[b,j] = 1/(1 + ||Z_b - SOMnew_j||)   via WMMA GEMM + fused epilogue
// ---------------------------------------------------------------------------
__global__ void gemm_q_kernel(const float* __restrict__ Z, const float* __restrict__ Sn,
                              const float* __restrict__ znorm, const float* __restrict__ s2n,
                              float* __restrict__ Qraw) {
  int wave = (int)(blockIdx.x * blockDim.x + threadIdx.x) >> 5;
  int lane = threadIdx.x & 31;
  int tn = wave & ((LAT / 16) - 1);
  int tm = wave >> 6;
  int m0 = tm * 16, n0 = tn * 16;
  v8f acc = gemm_tile_rowmajor(Z, Sn, m0, n0, lane, EMB);
  int n  = n0 + (lane & 15);
  int mb = m0 + ((lane >> 4) << 3);
  float sn = s2n[n];
#pragma unroll
  for (int i = 0; i < 8; ++i) {
    int m = mb + i;
    float dsq = znorm[m] - 2.0f * acc[i] + sn;
    float d = sqrtf(fmaxf(dsq, 0.0f));
    Qraw[(size_t)m * LAT + n] = 1.0f / (1.0f + d);
  }
}

// ---------------------------------------------------------------------------
// K8: qsumrow[b] = sum_j Qraw[b,j]
// ---------------------------------------------------------------------------
__global__ void rowsum_kernel(const float* __restrict__ in, float* __restrict__ out,
                              int rows) {
  int row = (int)(blockIdx.x * blockDim.x + threadIdx.x) >> 5;
  int lane = threadIdx.x & 31;
  if (row >= rows) return;
  const float* p = in + (size_t)row * LAT;
  float s = 0.0f;
#pragma unroll
  for (int c = lane; c < LAT; c += 32) s += p[c];
  s = wred_add(s);
  if (lane == 0) out[row] = s;
}

// ---------------------------------------------------------------------------
// K9: colsum[j] = sum_b Qraw[b,j]/qsumrow[b]   (= sum_b q[b,j])
// ---------------------------------------------------------------------------
__global__ void qcolsum_kernel(const float* __restrict__ Qraw, const float* __restrict__ qsr,
                               float* __restrict__ out) {
  int j = (int)(blockIdx.x * blockDim.x + threadIdx.x) >> 5;
  int lane = threadIdx.x & 31;
  if (j >= LAT) return;
  float s = 0.0f;
#pragma unroll
  for (int b = lane; b < BATCH; b += 32)
    s += Qraw[(size_t)b * LAT + j] / qsr[b];
  s = wred_add(s);
  if (lane == 0) out[j] = s;
}

// ---------------------------------------------------------------------------
// K10: kl[b] = sum_l p*log(p/q);  p = (q^2/colsum)/psum_row
// ---------------------------------------------------------------------------
__global__ void kl_kernel(const float* __restrict__ Qraw, const float* __restrict__ qsr,
                          const float* __restrict__ csum, float* __restrict__ kl) {
  int b = (int)(blockIdx.x * blockDim.x + threadIdx.x) >> 5;
  int lane = threadIdx.x & 31;
  if (b >= BATCH) return;
  float rq = 1.0f / qsr[b];
  const float* row = Qraw + (size_t)b * LAT;
  float psum = 0.0f;
#pragma unroll
  for (int l = lane; l < LAT; l += 32) {
    float q = row[l] * rq;
    psum += q * q / csum[l];
  }
  psum = wred_add(psum);              // all lanes hold total
  float acc = 0.0f;
#pragma unroll
  for (int l = lane; l < LAT; l += 32) {
    float q = row[l] * rq;
    float rp = q * q / csum[l];
    float p = rp / psum;
    acc += p * logf(p / q);
  }
  acc = wred_add(acc);
  if (lane == 0) kl[b] = acc;
}

// ---------------------------------------------------------------------------
// Host side
// ---------------------------------------------------------------------------
extern "C" void kernel_launch(void* const* d_in, const int* in_sizes, int n_in,
                              void* d_out, int out_size, void* d_ws, size_t ws_size,
                              hipStream_t stream) {
  const float* Z   = (const float*)d_in[0];   // [512,256]
  const float* SOM = (const float*)d_in[1];   // [32,32,256] -> [1024,256]
  const int*   stp = (const int*)d_in[2];     // scalar step
  float* out = (float*)d_out;                 // kl[512] ++ delta[512*1024*256]

  float* ws = (float*)d_ws;
  // workspace layout (float offsets), 64-float aligned
  const size_t OFF_PARAMS = 0;                      // 8
  const size_t OFF_ZNORM  = 64;                     // 512
  const size_t OFF_SNORM  = 576;                    // 1024
  const size_t OFF_S2N    = 1600;                   // 1024
  const size_t OFF_HSUM   = 2624;                   // 1024
  const size_t OFF_QSR    = 3648;                   // 512
  const size_t OFF_CSUM   = 4160;                   // 1024
  const size_t OFF_U      = 5184;                   // 512 ints
  const size_t OFF_DS     = 5696;                   // 512*1024 score; later Qraw
  const size_t OFF_HB     = OFF_DS + (size_t)BATCH * LAT;   // 512*1024 h-table
  const size_t OFF_SN     = OFF_HB + (size_t)BATCH * LAT;   // 1024*256 SOM_new
  // total: ~5.3 MB of d_ws

  params_kernel<<<1, 1, 0, stream>>>(stp, ws + OFF_PARAMS);
  rownorm_kernel<<<BATCH / 8, 256, 0, stream>>>(Z, ws + OFF_ZNORM, BATCH);
  rownorm_kernel<<<LAT / 8, 256, 0, stream>>>(SOM, ws + OFF_SNORM, LAT);

  // score GEMM: (512/16)*(1024/16)=2048 waves, 8 waves/block
  gemm_score_kernel<<<2048 / 8, 256, 0, stream>>>(Z, SOM, ws + OFF_SNORM, ws + OFF_DS);
  argmin_kernel<<<BATCH / 8, 256, 0, stream>>>(ws + OFF_DS, (int*)(ws + OFF_U));
  hbuf_kernel<<<(BATCH * LAT) / 256, 256, 0, stream>>>((const int*)(ws + OFF_U),
                                                       ws + OFF_PARAMS, ws + OFF_HB);

  // the 536 MB streaming writer (dominant cost, HBM store-bound, NT stores)
  delta_kernel<<<(BATCH * LAT * EMB / 4) / 256, 256, 0, stream>>>(
      Z, SOM, ws + OFF_HB, ws + OFF_PARAMS, out + BATCH);

  hsum_kernel<<<LAT / 8, 256, 0, stream>>>(ws + OFF_HB, ws + OFF_HSUM);
  // H^T Z GEMM: (1024/16)*(256/16)=1024 waves
  gemm_hz_kernel<<<1024 / 8, 256, 0, stream>>>(ws + OFF_HB, Z, SOM, ws + OFF_HSUM,
                                               ws + OFF_PARAMS, ws + OFF_SN);
  rownorm_kernel<<<LAT / 8, 256, 0, stream>>>(ws + OFF_SN, ws + OFF_S2N, LAT);

  // KL distance GEMM (Qraw aliases dead score buffer)
  gemm_q_kernel<<<2048 / 8, 256, 0, stream>>>(Z, ws + OFF_SN, ws + OFF_ZNORM,
                                              ws + OFF_S2N, ws + OFF_DS);
  rowsum_kernel<<<BATCH / 8, 256, 0, stream>>>(ws + OFF_DS, ws + OFF_QSR, BATCH);
  qcolsum_kernel<<<LAT / 8, 256, 0, stream>>>(ws + OFF_DS, ws + OFF_QSR, ws + OFF_CSUM);
  kl_kernel<<<BATCH / 8, 256, 0, stream>>>(ws + OFF_DS, ws + OFF_QSR, ws + OFF_CSUM, out);
}